// MarketStock_CrossAttention_57329223467620
// MI455X (gfx1250) — compile-verified
//
#include <hip/hip_runtime.h>
#include <hip/hip_bf16.h>

// ============================================================================
// MarketStock cross-attention for MI455X (gfx1250, wave32, WMMA).
// All GEMM stages use V_WMMA_F32_16X16X4_F32 (exact f32 semantics, matches
// the f32 reference). Dominant cost: diffusion GEMM A[4096x4096] x q'[4096x2048]
// (~69 GFLOP). A is materialized once (67 MB -> L2-resident on 192MB L2).
// The diffusion GEMM uses a double-buffered GLOBAL_LOAD_ASYNC_TO_LDS pipeline
// (ASYNCcnt-tracked) when the toolchain exposes the builtin.
//
// Workspace requirement: ~192 MiB (see offsets in kernel_launch).
// ============================================================================

typedef float v2f __attribute__((ext_vector_type(2)));
typedef float v8f __attribute__((ext_vector_type(8)));

#define NNODE 4096   // B = N_NODE + 1
#define TT    32     // T
#define DM    64     // D_MODEL
#define TD    2048   // T * D_MODEL
#define DE    32     // D_EMB

// ---------------- async global->LDS copy (CDNA5) ----------------------------
// Builtin prototype (from hipcc diagnostic):
//   param0: int4 (vector_size 16) pointer, AS1 (__device__), non-const
//   param1: int4 pointer, AS3 (__shared__)
//   param2: imm offset, param3: cpol
#if __has_builtin(__builtin_amdgcn_global_load_async_to_lds_b128)
#define USE_ASYNC_LDS 1
#if __has_builtin(__builtin_amdgcn_s_wait_asynccnt)
#define WAIT_ASYNC(n) __builtin_amdgcn_s_wait_asynccnt(n)
#else
#define WAIT_ASYNC(n) asm volatile("s_wait_asynccnt %0" ::"n"(n) : "memory")
#endif
typedef int async_v4i __attribute__((vector_size(16)));
typedef __attribute__((address_space(1))) async_v4i* gas_p;
typedef __attribute__((address_space(3))) async_v4i* las_p;
static __device__ __forceinline__ void async_cp16(const float* g, float* l) {
  __builtin_amdgcn_global_load_async_to_lds_b128((gas_p)g, (las_p)l, 0, 0);
}
#else
#define USE_ASYNC_LDS 0
#endif

// D = A(16x4 f32) * B(4x16 f32) + C(16x16 f32).  wave32, EXEC must be all 1s.
static __device__ __forceinline__ v8f wmma4(v2f a, v2f b, v8f c) {
  return __builtin_amdgcn_wmma_f32_16x16x4_f32(
      /*neg_a=*/false, a, /*neg_b=*/false, b,
      /*c_mod=*/(short)0, c, /*reuse_a=*/false, /*reuse_b=*/false);
}

static __device__ __forceinline__ float elu1(float x) {
  return x > 0.f ? x : (expf(x) - 1.f);
}

// ----------------------------------------------------------------------------
// Y[r, o] = sum_i X[r, i] * W[o, i] + bias[o]   (torch Linear: x @ W.T + b)
// 1024 blocks of 256 threads; each block: 128 rows, 8 waves x (16 rows x 64 cols).
// W is staged transposed in LDS so the inner loop is pure ds_load + wmma.
// ----------------------------------------------------------------------------
__global__ __launch_bounds__(256) void proj_kernel(
    const float* __restrict__ X, const float* __restrict__ W,
    const float* __restrict__ bias, float* __restrict__ Y) {
  __shared__ float Xs[128 * 68];  // stride 68 -> float4-aligned, conflict-free
  __shared__ float Ws[64 * 68];   // Ws[k][o] = W[o*64 + k]  (W.T)
  const int tid = threadIdx.x;
  const int R0 = blockIdx.x * 128;
#pragma unroll
  for (int jj = 0; jj < 8; ++jj) {
    int linear = jj * 256 + tid;
    int r = linear >> 4, c4 = linear & 15;
    *reinterpret_cast<float4*>(&Xs[r * 68 + c4 * 4]) =
        *reinterpret_cast<const float4*>(&X[(size_t)(R0 + r) * DM + c4 * 4]);
  }
#pragma unroll
  for (int jj = 0; jj < 16; ++jj) {  // 4096 floats of W, transposed on store
    int f = jj * 256 + tid;          // f = o*64 + k (coalesced read)
    Ws[(f & 63) * 68 + (f >> 6)] = W[f];
  }
  __syncthreads();
  const int wave = tid >> 5, lane = tid & 31;
  const int half = lane >> 4, ln = lane & 15, koff = half * 2;
  const int rbase = wave * 16;
  v8f acc[4] = {};
  for (int kb = 0; kb < DM; kb += 4) {
    v2f a;
    a.x = Xs[(rbase + ln) * 68 + kb + koff];
    a.y = Xs[(rbase + ln) * 68 + kb + koff + 1];
#pragma unroll
    for (int tc = 0; tc < 4; ++tc) {
      v2f b;
      b.x = Ws[(kb + koff) * 68 + tc * 16 + ln];
      b.y = Ws[(kb + koff + 1) * 68 + tc * 16 + ln];
      acc[tc] = wmma4(a, b, acc[tc]);
    }
  }
#pragma unroll
  for (int tc = 0; tc < 4; ++tc) {
    int col = tc * 16 + ln;
    float bb = bias[col];
#pragma unroll
    for (int j = 0; j < 8; ++j) {
      int row = R0 + rbase + j + half * 8;
      Y[(size_t)row * DM + col] = acc[tc][j] + bb;
    }
  }
}

// ----------------------------------------------------------------------------
// s[n] = sum_e emb[n, e]
// ----------------------------------------------------------------------------
__global__ __launch_bounds__(256) void rowsum_emb_kernel(
    const float* __restrict__ emb, float* __restrict__ s) {
  int n = blockIdx.x * 256 + threadIdx.x;
  if (n < NNODE) {
    float a = 0.f;
#pragma unroll
    for (int e = 0; e < DE; ++e) a += emb[n * DE + e];
    s[n] = a;
  }
}

// ----------------------------------------------------------------------------
// rZ[n] = 1 / sum_m exp(elu(s[n]-s[m]))   (one wave per row; shfl reduce)
// ----------------------------------------------------------------------------
__global__ __launch_bounds__(256) void softmax_denom_kernel(
    const float* __restrict__ s, float* __restrict__ rZ) {
  int wave = threadIdx.x >> 5, lane = threadIdx.x & 31;
  int n = blockIdx.x * 8 + wave;
  float sn = s[n];
  float p = 0.f;
  for (int m = lane; m < NNODE; m += 32) p += expf(elu1(sn - s[m]));
#pragma unroll
  for (int off = 16; off > 0; off >>= 1) p += __shfl_down(p, off, 32);
  if (lane == 0) rZ[n] = 1.f / p;
}

// ----------------------------------------------------------------------------
// A[n, m] = exp(elu(s[n]-s[m])) * rZ[n] + (n==m)
// ----------------------------------------------------------------------------
__global__ __launch_bounds__(256) void build_A_kernel(
    const float* __restrict__ s, const float* __restrict__ rZ,
    float* __restrict__ A) {
  int n = blockIdx.x;
  float sn = s[n], r = rZ[n];
  for (int m = threadIdx.x; m < NNODE; m += 256) {
    float val = expf(elu1(sn - s[m])) * r;
    if (m == n) val += 1.f;
    A[(size_t)n * NNODE + m] = val;
  }
}

// ----------------------------------------------------------------------------
// C[n, td] = sum_m A[n, m] * B[m, td]   (M=4096, K=4096, N=2048)
// Block tile 128x64, 8 waves in a 4x2 grid, each wave 32x32 (2x2 WMMA accs).
// K staged 32-wide in double-buffered LDS via async global->LDS copies
// (6 x b128 per wave per stage: 4 for A-tile, 2 for B-tile).
// ----------------------------------------------------------------------------
__global__ __launch_bounds__(256) void diffusion_gemm_kernel(
    const float* __restrict__ A, const float* __restrict__ B,
    float* __restrict__ C) {
  __shared__ float As[2][128 * 36];  // 2 x 18.4 KB
  __shared__ float Bs[2][32 * 68];   // 2 x  8.7 KB
  const int tid = threadIdx.x;
  const int M0 = blockIdx.y * 128, N0 = blockIdx.x * 64;
  const int wave = tid >> 5, lane = tid & 31;
  const int half = lane >> 4, ln = lane & 15, koff = half * 2;
  const int wm = wave >> 1, wn = wave & 1;
  const int rb = wm * 32, cb = wn * 32;

#if USE_ASYNC_LDS
  // Each wave owns 4 A-issues (128 f4) + 2 B-issues (64 f4) per stage.
  auto issue_stage = [&](int kb0, int buf) {
#pragma unroll
    for (int j = 0; j < 4; ++j) {
      int lin = wave * 128 + j * 32 + lane;
      int r = lin >> 3, c4 = lin & 7;
      async_cp16(&A[(size_t)(M0 + r) * NNODE + kb0 + c4 * 4],
                 &As[buf][r * 36 + c4 * 4]);
    }
#pragma unroll
    for (int j = 0; j < 2; ++j) {
      int lin = wave * 64 + j * 32 + lane;
      int kk = lin >> 4, c4 = lin & 15;
      async_cp16(&B[(size_t)(kb0 + kk) * TD + N0 + c4 * 4],
                 &Bs[buf][kk * 68 + c4 * 4]);
    }
  };
#else
  auto issue_stage = [&](int kb0, int buf) {
#pragma unroll
    for (int jj = 0; jj < 4; ++jj) {
      int linear = jj * 256 + tid;
      int r = linear >> 3, c4 = linear & 7;
      *reinterpret_cast<float4*>(&As[buf][r * 36 + c4 * 4]) =
          *reinterpret_cast<const float4*>(
              &A[(size_t)(M0 + r) * NNODE + kb0 + c4 * 4]);
    }
#pragma unroll
    for (int jj = 0; jj < 2; ++jj) {
      int linear = jj * 256 + tid;
      int kk = linear >> 4, c4 = linear & 15;
      *reinterpret_cast<float4*>(&Bs[buf][kk * 68 + c4 * 4]) =
          *reinterpret_cast<const float4*>(
              &B[(size_t)(kb0 + kk) * TD + N0 + c4 * 4]);
    }
  };
#endif

  v8f acc[2][2] = {};
  issue_stage(0, 0);
  const int NSTAGE = NNODE / 32;  // 128
  for (int s = 0; s < NSTAGE; ++s) {
    const int cur = s & 1;
    if (s + 1 < NSTAGE) issue_stage((s + 1) * 32, cur ^ 1);
#if USE_ASYNC_LDS
    if (s + 1 < NSTAGE) {
      WAIT_ASYNC(6);  // stage s complete; stage s+1 (6 issues) may be in flight
    } else {
      WAIT_ASYNC(0);
    }
#endif
    __syncthreads();  // all waves' tiles for stage s visible
    const float* Asc = As[cur];
    const float* Bsc = Bs[cur];
#pragma unroll
    for (int ks = 0; ks < 8; ++ks) {
      int kb = ks * 4;
      v2f a0, a1, b0, b1;
      a0.x = Asc[(rb + ln) * 36 + kb + koff];
      a0.y = Asc[(rb + ln) * 36 + kb + koff + 1];
      a1.x = Asc[(rb + 16 + ln) * 36 + kb + koff];
      a1.y = Asc[(rb + 16 + ln) * 36 + kb + koff + 1];
      b0.x = Bsc[(kb + koff) * 68 + cb + ln];
      b0.y = Bsc[(kb + koff + 1) * 68 + cb + ln];
      b1.x = Bsc[(kb + koff) * 68 + cb + 16 + ln];
      b1.y = Bsc[(kb + koff + 1) * 68 + cb + 16 + ln];
      acc[0][0] = wmma4(a0, b0, acc[0][0]);
      acc[0][1] = wmma4(a0, b1, acc[0][1]);
      acc[1][0] = wmma4(a1, b0, acc[1][0]);
      acc[1][1] = wmma4(a1, b1, acc[1][1]);
    }
    __syncthreads();  // done reading buf[cur] before it is overwritten
  }
#pragma unroll
  for (int ti = 0; ti < 2; ++ti)
#pragma unroll
    for (int tj = 0; tj < 2; ++tj) {
      int col = N0 + cb + tj * 16 + ln;
#pragma unroll
      for (int j = 0; j < 8; ++j) {
        int row = M0 + rb + ti * 16 + j + half * 8;
        C[(size_t)row * TD + col] = acc[ti][tj][j];
      }
    }
}

// ----------------------------------------------------------------------------
// Per node n:  Wn = emb[n] . We  ([64,64]),  bn = emb[n] . be,
//              qf[n] = qd[n] ([32,64]) @ Wn + bn
// ----------------------------------------------------------------------------
__global__ __launch_bounds__(256) void node_transform_kernel(
    const float* __restrict__ qd, const float* __restrict__ emb,
    const float* __restrict__ We, const float* __restrict__ be,
    float* __restrict__ qf) {
  __shared__ float Wn[64 * 68];
  __shared__ float Xs[32 * 68];
  __shared__ float bn[DM];
  __shared__ float embn[DE];
  const int tid = threadIdx.x;
  const int n = blockIdx.x;
  if (tid < DE) embn[tid] = emb[n * DE + tid];
#pragma unroll
  for (int jj = 0; jj < 8; ++jj) {
    int linear = jj * 256 + tid;
    Xs[(linear >> 6) * 68 + (linear & 63)] = qd[(size_t)n * TD + linear];
  }
  __syncthreads();
#pragma unroll
  for (int jj = 0; jj < 16; ++jj) {  // Wn[i][o], f = i*64+o
    int f = jj * 256 + tid;
    float a = 0.f;
#pragma unroll
    for (int e = 0; e < DE; ++e) a += embn[e] * We[e * 4096 + f];
    Wn[(f >> 6) * 68 + (f & 63)] = a;
  }
  if (tid < DM) {
    float a = 0.f;
#pragma unroll
    for (int e = 0; e < DE; ++e) a += embn[e] * be[e * DM + tid];
    bn[tid] = a;
  }
  __syncthreads();
  const int wave = tid >> 5, lane = tid & 31;
  const int half = lane >> 4, ln = lane & 15, koff = half * 2;
  const int tr = wave >> 2, tc = wave & 3;  // 2x4 tile grid over [32,64]
  v8f acc = {};
#pragma unroll
  for (int kb = 0; kb < DM; kb += 4) {
    v2f a, b;
    a.x = Xs[(tr * 16 + ln) * 68 + kb + koff];
    a.y = Xs[(tr * 16 + ln) * 68 + kb + koff + 1];
    b.x = Wn[(kb + koff) * 68 + tc * 16 + ln];
    b.y = Wn[(kb + koff + 1) * 68 + tc * 16 + ln];
    acc = wmma4(a, b, acc);
  }
  int col = tc * 16 + ln;
  float bb = bn[col];
#pragma unroll
  for (int j = 0; j < 8; ++j) {
    int t = tr * 16 + j + half * 8;
    qf[(size_t)n * TD + t * DM + col] = acc[j] + bb;
  }
}

// ----------------------------------------------------------------------------
// Per node n: 8-head attention (head h = dims [8h, 8h+8)), scores 32x32,
// softmax over s, out = attn @ v; then y = out @ Wo.T + bo via WMMA.
// One wave per head for phase 1 (thread = (h, t)).
// ----------------------------------------------------------------------------
__global__ __launch_bounds__(256) void attention_kernel(
    const float* __restrict__ qf, const float* __restrict__ kp,
    const float* __restrict__ vp, const float* __restrict__ Wo,
    const float* __restrict__ bo, float* __restrict__ out) {
  __shared__ float Ks[32 * 68];
  __shared__ float Vs[32 * 68];
  __shared__ float Os[32 * 68];
  const int tid = threadIdx.x;
  const int n = blockIdx.x;
#pragma unroll
  for (int jj = 0; jj < 8; ++jj) {
    int linear = jj * 256 + tid;
    int r = linear >> 6, c = linear & 63;
    Ks[r * 68 + c] = kp[(size_t)n * TD + linear];
    Vs[r * 68 + c] = vp[(size_t)n * TD + linear];
  }
  __syncthreads();
  const int h = tid >> 5, t = tid & 31;
  const int h8 = h * 8;
  float q8[8];
#pragma unroll
  for (int d = 0; d < 8; ++d) q8[d] = qf[(size_t)n * TD + t * DM + h8 + d];
  float sc[32];
  float mx = -3.0e38f;
#pragma unroll
  for (int ss = 0; ss < 32; ++ss) {
    float dot = 0.f;
#pragma unroll
    for (int d = 0; d < 8; ++d) dot += q8[d] * Ks[ss * 68 + h8 + d];
    dot *= 0.35355339059327373f;  // 1/sqrt(D/h)
    sc[ss] = dot;
    mx = fmaxf(mx, dot);
  }
  float sum = 0.f;
#pragma unroll
  for (int ss = 0; ss < 32; ++ss) {
    sc[ss] = expf(sc[ss] - mx);
    sum += sc[ss];
  }
  float rinv = 1.f / sum;
  float av[8] = {};
#pragma unroll
  for (int ss = 0; ss < 32; ++ss) {
    float w = sc[ss] * rinv;
#pragma unroll
    for (int d = 0; d < 8; ++d) av[d] += w * Vs[ss * 68 + h8 + d];
  }
#pragma unroll
  for (int d = 0; d < 8; ++d) Os[t * 68 + h8 + d] = av[d];
  __syncthreads();
  const int wave = tid >> 5, lane = tid & 31;
  const int half = lane >> 4, ln = lane & 15, koff = half * 2;
  const int tr = wave >> 2, tc = wave & 3;
  v8f acc = {};
#pragma unroll
  for (int kb = 0; kb < DM; kb += 4) {
    v2f a, b;
    a.x = Os[(tr * 16 + ln) * 68 + kb + koff];
    a.y = Os[(tr * 16 + ln) * 68 + kb + koff + 1];
    b.x = Wo[(tc * 16 + ln) * DM + kb + koff];  // B[k=d,n=o] = Wo[o*64+d]
    b.y = Wo[(tc * 16 + ln) * DM + kb + koff + 1];
    acc = wmma4(a, b, acc);
  }
  int col = tc * 16 + ln;
  float bb = bo[col];
#pragma unroll
  for (int j = 0; j < 8; ++j) {
    int tt = tr * 16 + j + half * 8;
    out[(size_t)n * TD + tt * DM + col] = acc[j] + bb;
  }
}

// ============================================================================
extern "C" void kernel_launch(void* const* d_in, const int* in_sizes, int n_in,
                              void* d_out, int out_size, void* d_ws,
                              size_t ws_size, hipStream_t stream) {
  (void)in_sizes; (void)n_in; (void)out_size; (void)ws_size;
  const float* q   = (const float*)d_in[0];
  const float* k   = (const float*)d_in[1];
  const float* v   = (const float*)d_in[2];
  const float* emb = (const float*)d_in[3];
  const float* We  = (const float*)d_in[4];
  const float* be  = (const float*)d_in[5];
  const float* Wq  = (const float*)d_in[6];
  const float* bq  = (const float*)d_in[7];
  const float* Wk  = (const float*)d_in[8];
  const float* bk  = (const float*)d_in[9];
  const float* Wv  = (const float*)d_in[10];
  const float* bv  = (const float*)d_in[11];
  const float* Wo  = (const float*)d_in[12];
  const float* bo  = (const float*)d_in[13];
  float* out = (float*)d_out;

  // Workspace layout (bytes). Total need: ~192.1 MiB.
  char* ws = (char*)d_ws;
  float* qp   = (float*)(ws + 0);          // [4096,2048]  33.55 MB
  float* kp   = (float*)(ws + 33554432);   // [4096,2048]  33.55 MB
  float* vp   = (float*)(ws + 67108864);   // [4096,2048]  33.55 MB
  float* Amat = (float*)(ws + 100663296);  // [4096,4096]  67.11 MB
  float* qd   = (float*)(ws + 167772160);  // [4096,2048]  33.55 MB
  float* sv   = (float*)(ws + 201326592);  // [4096]
  float* rZ   = (float*)(ws + 201342976);  // [4096]
  float* qf   = qp;  // reuse: qp is dead after the diffusion GEMM

  proj_kernel<<<1024, 256, 0, stream>>>(q, Wq, bq, qp);
  proj_kernel<<<1024, 256, 0, stream>>>(k, Wk, bk, kp);
  proj_kernel<<<1024, 256, 0, stream>>>(v, Wv, bv, vp);
  rowsum_emb_kernel<<<16, 256, 0, stream>>>(emb, sv);
  softmax_denom_kernel<<<512, 256, 0, stream>>>(sv, rZ);
  build_A_kernel<<<4096, 256, 0, stream>>>(sv, rZ, Amat);
  diffusion_gemm_kernel<<<dim3(32, 32), 256, 0, stream>>>(Amat, qp, qd);
  node_transform_kernel<<<4096, 256, 0, stream>>>(qd, emb, We, be, qf);
  attention_kernel<<<4096, 256, 0, stream>>>(qf, kp, vp, Wo, bo, out);
}